// BiT_MamSleep_49306224558545
// MI455X (gfx1250) — compile-verified
//
#include <hip/hip_runtime.h>
#include <math.h>

#define HID     128
#define BATCH   4
#define SEQ     2048
#define D_STATE 16
#define D_CONV  4
#define D_INNER 256
#define DT_RANK 8
#define BL      (BATCH*SEQ)   // 8192 positions

typedef __attribute__((ext_vector_type(2))) float v2f;
typedef __attribute__((ext_vector_type(8))) float v8f;

__device__ __forceinline__ float silu_f(float x) { return x / (1.0f + __expf(-x)); }

// ---------------------------------------------------------------------------
// Generic GEMM: C[BL x N] = act( A[BL x K] @ W[N x K]^T + bias )
// One wave computes a 16x32 tile (two 16x16 WMMA accumulators sharing the A
// fragment) with V_WMMA_F32_16X16X4_F32.
// A-lane layout (32-bit 16x4): lanes0-15 K={kb+0,kb+1}, lanes16-31 K={kb+2,kb+3}
// revA=1 reverses the sequence (row l -> L-1-l within each batch).
// ACT: 0=none, 1=silu, 2=softplus.  MASKED: N not a multiple of 32.
// ---------------------------------------------------------------------------
template<bool MASKED, int ACT>
__global__ void __launch_bounds__(128)
k_gemm(const float* __restrict__ A, int ldA,
       const float* __restrict__ W,
       const float* __restrict__ bias,
       float* __restrict__ C, int ldC,
       int K, int Nact, int revA)
{
    const int lane = threadIdx.x & 31;
    const int wave = threadIdx.x >> 5;
    const int half = lane >> 4;     // which half-wave
    const int lid  = lane & 15;
    const int m0 = (blockIdx.x * 4 + wave) * 16;
    const int n0 = blockIdx.y * 32;

    int arow = m0 + lid;
    if (revA) { int b = arow / SEQ, l = arow % SEQ; arow = b * SEQ + (SEQ - 1 - l); }
    const float* __restrict__ Ap = A + (size_t)arow * ldA + 2 * half;

    const int nc0 = n0 + lid;
    const int nc1 = n0 + 16 + lid;
    const float m0f = (!MASKED || nc0 < Nact) ? 1.0f : 0.0f;
    const float m1f = (!MASKED || nc1 < Nact) ? 1.0f : 0.0f;
    const float* __restrict__ Wp0 =
        W + (size_t)((!MASKED || nc0 < Nact) ? nc0 : 0) * K + 2 * half;
    const float* __restrict__ Wp1 =
        W + (size_t)((!MASKED || nc1 < Nact) ? nc1 : 0) * K + 2 * half;

    v8f acc0 = {}, acc1 = {};
    for (int kb = 0; kb < K; kb += 4) {
        v2f av, b0, b1;
        av.x = Ap[kb];
        av.y = Ap[kb + 1];
        b0.x = Wp0[kb];     b0.y = Wp0[kb + 1];
        b1.x = Wp1[kb];     b1.y = Wp1[kb + 1];
        if (MASKED) {                       // compile-time: only xp GEMM pays this
            b0.x *= m0f; b0.y *= m0f;
            b1.x *= m1f; b1.y *= m1f;
        }
        acc0 = __builtin_amdgcn_wmma_f32_16x16x4_f32(
                   false, av, false, b0, (short)0, acc0, false, false);
        acc1 = __builtin_amdgcn_wmma_f32_16x16x4_f32(
                   false, av, false, b1, (short)0, acc1, false, false);
    }

    const int mrow = m0 + half * 8;   // VGPR r: lanes0-15 -> M=r, lanes16-31 -> M=r+8
    #pragma unroll 2
    for (int t = 0; t < 2; ++t) {
        const int ncol = t ? nc1 : nc0;
        const v8f& acc = t ? acc1 : acc0;
        if (!MASKED || ncol < Nact) {
            const float bb = bias ? bias[ncol] : 0.0f;
            #pragma unroll
            for (int r = 0; r < 8; ++r) {
                float v = acc[r] + bb;
                if (ACT == 1)      v = silu_f(v);
                else if (ACT == 2) v = (v > 20.0f) ? v : log1pf(__expf(v));
                C[(size_t)(mrow + r) * ldC + ncol] = v;
            }
        }
    }
}

// ---------------------------------------------------------------------------
// LayerNorm over H at each (b,l): input (B,H,L) strided -> output (B*L, H)
// one wave per position; 4 values per lane
// ---------------------------------------------------------------------------
__global__ void __launch_bounds__(256)
k_ln_in(const float* __restrict__ x, const float* __restrict__ g,
        const float* __restrict__ bta, float* __restrict__ xn)
{
    const int pos  = blockIdx.x * 8 + (threadIdx.x >> 5);
    const int lane = threadIdx.x & 31;
    const int b = pos / SEQ, l = pos % SEQ;
    const float* xp = x + (size_t)b * HID * SEQ + l;  // element h at xp[h*SEQ]
    float v[4], s = 0.f, s2 = 0.f;
    #pragma unroll
    for (int i = 0; i < 4; ++i) {
        int h = lane + 32 * i;
        v[i] = xp[(size_t)h * SEQ];
        s += v[i]; s2 += v[i] * v[i];
    }
    #pragma unroll
    for (int m = 16; m >= 1; m >>= 1) { s += __shfl_xor(s, m, 32); s2 += __shfl_xor(s2, m, 32); }
    const float mean = s * (1.0f / HID);
    const float var  = s2 * (1.0f / HID) - mean * mean;
    const float inv  = rsqrtf(var + 1e-5f);
    #pragma unroll
    for (int i = 0; i < 4; ++i) {
        int h = lane + 32 * i;
        xn[(size_t)pos * HID + h] = (v[i] - mean) * inv * g[h] + bta[h];
    }
}

// LayerNorm over H, input (B*L, H) contiguous -> output (B,H,L) strided
__global__ void __launch_bounds__(256)
k_ln_out(const float* __restrict__ t, const float* __restrict__ g,
         const float* __restrict__ bta, float* __restrict__ out)
{
    const int pos  = blockIdx.x * 8 + (threadIdx.x >> 5);
    const int lane = threadIdx.x & 31;
    const int b = pos / SEQ, l = pos % SEQ;
    const float* tp = t + (size_t)pos * HID;
    float v[4], s = 0.f, s2 = 0.f;
    #pragma unroll
    for (int i = 0; i < 4; ++i) {
        int h = lane + 32 * i;
        v[i] = tp[h];
        s += v[i]; s2 += v[i] * v[i];
    }
    #pragma unroll
    for (int m = 16; m >= 1; m >>= 1) { s += __shfl_xor(s, m, 32); s2 += __shfl_xor(s2, m, 32); }
    const float mean = s * (1.0f / HID);
    const float var  = s2 * (1.0f / HID) - mean * mean;
    const float inv  = rsqrtf(var + 1e-5f);
    #pragma unroll
    for (int i = 0; i < 4; ++i) {
        int h = lane + 32 * i;
        out[(size_t)b * HID * SEQ + (size_t)h * SEQ + l] = (v[i] - mean) * inv * g[h] + bta[h];
    }
}

// ---------------------------------------------------------------------------
// Depthwise causal conv (taps=4) + SiLU.  u = xz[:, 0:256] (ld 512) -> uc (ld 256)
// ---------------------------------------------------------------------------
__global__ void __launch_bounds__(256)
k_conv(const float* __restrict__ xz, const float* __restrict__ cw,
       const float* __restrict__ cb, float* __restrict__ uc)
{
    const int idx = blockIdx.x * 256 + threadIdx.x;        // over BL*D_INNER
    const int d  = idx & (D_INNER - 1);
    const int bl = idx / D_INNER;
    const int l  = bl % SEQ;
    const int b  = bl / SEQ;
    float s = cb[d];
    #pragma unroll
    for (int k = 0; k < D_CONV; ++k) {
        int ls = l + k - (D_CONV - 1);
        if (ls >= 0) s += cw[d * D_CONV + k] * xz[((size_t)(b * SEQ + ls)) * 512 + d];
    }
    uc[idx] = silu_f(s);
}

// ---------------------------------------------------------------------------
// Selective scan.  16-lane group per (b,d); lane = state s.  Sequential in L.
// Fuses  y = scan + uc*D  and  y2 = y * silu(z).
// ---------------------------------------------------------------------------
__global__ void __launch_bounds__(256)
k_scan(const float* __restrict__ dt, const float* __restrict__ dbl,
       const float* __restrict__ uc, const float* __restrict__ xz,
       const float* __restrict__ Alog, const float* __restrict__ Dp,
       float* __restrict__ y2)
{
    const int gid = (blockIdx.x * 256 + threadIdx.x) >> 4;  // pair index (b,d)
    const int s   = threadIdx.x & 15;
    const int b   = gid >> 8;           // / D_INNER
    const int d   = gid & (D_INNER - 1);
    const float A  = -__expf(Alog[d * D_STATE + s]);
    const float Dd = Dp[d];
    float h = 0.0f;
    for (int l = 0; l < SEQ; ++l) {
        const size_t pos = (size_t)(b * SEQ + l);
        const float dtv = dt [pos * D_INNER + d];
        const float Bv  = dbl[pos * 64 + DT_RANK + s];
        const float Cv  = dbl[pos * 64 + DT_RANK + D_STATE + s];
        const float uv  = uc [pos * D_INNER + d];
        h = __expf(dtv * A) * h + dtv * Bv * uv;
        float p = h * Cv;
        p += __shfl_xor(p, 1, 16);
        p += __shfl_xor(p, 2, 16);
        p += __shfl_xor(p, 4, 16);
        p += __shfl_xor(p, 8, 16);
        if (s == 0) {
            const float zv = xz[pos * 512 + D_INNER + d];
            y2[pos * D_INNER + d] = (p + uv * Dd) * (zv * (1.0f / (1.0f + __expf(-zv))));
        }
    }
}

// y_comb = (yf[b,l] + yb[b,L-1-l]) * gate[b,l]     (all (BL,H))
__global__ void __launch_bounds__(256)
k_combine(const float* __restrict__ yf, const float* __restrict__ yb,
          const float* __restrict__ gate, float* __restrict__ out)
{
    const int idx = blockIdx.x * 256 + threadIdx.x;        // over BL*HID
    const int h   = idx & (HID - 1);
    const int pos = idx / HID;
    const int l   = pos % SEQ;
    const int b   = pos / SEQ;
    const size_t rpos = (size_t)(b * SEQ + (SEQ - 1 - l));
    out[idx] = (yf[idx] + yb[rpos * HID + h]) * gate[idx];
}

// ---------------------------------------------------------------------------
static inline void gemm(hipStream_t st, const float* A, int ldA, const float* W,
                        const float* bias, float* C, int ldC,
                        int K, int N, int act, int revA)
{
    const int Nt  = (N + 15) / 16;
    const int Nt2 = (Nt + 1) / 2;             // 32 columns per wave
    dim3 grid(BL / 16 / 4, Nt2);
    dim3 blk(128);
    const bool masked = (N % 32) != 0;
    if (masked) {
        k_gemm<true, 0><<<grid, blk, 0, st>>>(A, ldA, W, bias, C, ldC, K, N, revA);
    } else if (act == 1) {
        k_gemm<false, 1><<<grid, blk, 0, st>>>(A, ldA, W, bias, C, ldC, K, N, revA);
    } else if (act == 2) {
        k_gemm<false, 2><<<grid, blk, 0, st>>>(A, ldA, W, bias, C, ldC, K, N, revA);
    } else {
        k_gemm<false, 0><<<grid, blk, 0, st>>>(A, ldA, W, bias, C, ldC, K, N, revA);
    }
}

extern "C" void kernel_launch(void* const* d_in, const int* in_sizes, int n_in,
                              void* d_out, int out_size, void* d_ws, size_t ws_size,
                              hipStream_t stream)
{
    const float* x    = (const float*)d_in[0];
    const float* ln_g = (const float*)d_in[1];
    const float* ln_b = (const float*)d_in[2];
    const float* lm_w = (const float*)d_in[3];
    const float* lm_b = (const float*)d_in[4];
    const float* cf_w = (const float*)d_in[5];
    const float* cf_b = (const float*)d_in[6];
    const float* cb_w = (const float*)d_in[7];
    const float* cb_b = (const float*)d_in[8];
    const float* lg_w = (const float*)d_in[9];
    const float* lg_b = (const float*)d_in[10];
    const float* lo_w = (const float*)d_in[11];
    const float* lo_b = (const float*)d_in[12];
    // mamba params: [dir][0]=in_w [1]=conv_w [2]=conv_b [3]=xp_w [4]=dt_w
    //               [5]=dt_b [6]=Alog [7]=D [8]=out_w
    const float* mp[2][9];
    for (int dir = 0; dir < 2; ++dir)
        for (int i = 0; i < 9; ++i)
            mp[dir][i] = (const float*)d_in[13 + dir * 9 + i];

    float* ws = (float*)d_ws;
    float* xn   = ws; ws += (size_t)BL * HID;       // (BL,H) ; reused for y_comb
    float* t1   = ws; ws += (size_t)BL * HID;       // lm out ; reused for t2
    float* xmf  = ws; ws += (size_t)BL * HID;
    float* xmb  = ws; ws += (size_t)BL * HID;
    float* gate = ws; ws += (size_t)BL * HID;
    float* yf   = ws; ws += (size_t)BL * HID;
    float* yb   = ws; ws += (size_t)BL * HID;
    float* xz   = ws; ws += (size_t)BL * 512;       // per-direction, reused
    float* ucb  = ws; ws += (size_t)BL * D_INNER;
    float* dbl  = ws; ws += (size_t)BL * 64;        // ld 64, cols 0..39 valid
    float* dtb  = ws; ws += (size_t)BL * D_INNER;
    float* y2   = ws; ws += (size_t)BL * D_INNER;

    // 1. input LayerNorm: (B,H,L) -> (BL,H)
    k_ln_in<<<BL / 8, 256, 0, stream>>>(x, ln_g, ln_b, xn);

    // 2. channel mixes
    gemm(stream, xn, HID, lm_w, lm_b, t1,   HID, HID, HID, /*act*/0, 0);   // t1 = lm(xn)
    gemm(stream, t1, HID, cf_w, cf_b, xmf,  HID, HID, HID, /*silu*/1, 0);  // forward branch in
    gemm(stream, t1, HID, cb_w, cb_b, xmb,  HID, HID, HID, /*silu*/1, 1);  // reversed rows
    gemm(stream, xn, HID, lg_w, lg_b, gate, HID, HID, HID, /*silu*/1, 0);  // gate

    // 3. the two mamba branches
    for (int dir = 0; dir < 2; ++dir) {
        const float* xin  = dir ? xmb : xmf;
        float*       ydir = dir ? yb  : yf;
        const float** p = mp[dir];
        // in_proj -> xz (BL,512)
        gemm(stream, xin, HID, p[0], nullptr, xz, 512, HID, 512, 0, 0);
        // depthwise conv + silu -> uc
        k_conv<<<(BL * D_INNER) / 256, 256, 0, stream>>>(xz, p[1], p[2], ucb);
        // xp: uc @ xp_w^T -> dbl (BL,40 in ld64)  [masked path]
        gemm(stream, ucb, D_INNER, p[3], nullptr, dbl, 64, D_INNER,
             DT_RANK + 2 * D_STATE, 0, 0);
        // dt: softplus(dbl[:,:8] @ dt_w^T + dt_b) -> (BL,256)
        gemm(stream, dbl, 64, p[4], p[5], dtb, D_INNER, DT_RANK, D_INNER, /*softplus*/2, 0);
        // selective scan (+ uc*D, * silu(z)) -> y2
        k_scan<<<(BATCH * D_INNER * D_STATE) / 256, 256, 0, stream>>>(
            dtb, dbl, ucb, xz, p[6], p[7], y2);
        // out_proj -> ydir (BL,H)
        gemm(stream, y2, D_INNER, p[8], nullptr, ydir, HID, D_INNER, HID, 0, 0);
    }

    // 4. combine: (yf + reverse(yb)) * gate  -> reuse xn
    k_combine<<<(BL * HID) / 256, 256, 0, stream>>>(yf, yb, gate, xn);

    // 5. output projection -> reuse t1
    gemm(stream, xn, HID, lo_w, lo_b, t1, HID, HID, HID, 0, 0);

    // 6. final LayerNorm, write (B,H,L)
    k_ln_out<<<BL / 8, 256, 0, stream>>>(t1, ln_g, ln_b, (float*)d_out);
}